// PagedKVCache_83949430768212
// MI455X (gfx1250) — compile-verified
//
#include <hip/hip_runtime.h>
#include <hip/hip_bf16.h>

// ---------------------------------------------------------------------------
// PagedKVCache prefill write for MI455X (gfx1250).
//
// Pure bandwidth problem (~576 MiB => ~25us @ 23.3 TB/s). No matmul -> no
// WMMA. CDNA5-specific paths used instead:
//   * bulk cache copy via async global<->LDS DMA (ASYNCcnt-tracked
//     global_load_async_to_lds_b128 / global_store_async_from_lds_b128,
//     fenced by s_wait_asynccnt) -- data never occupies VGPRs.
//   * non-temporal B128 loads/stores for the scatter (streams > 192MB L2).
// ---------------------------------------------------------------------------

// Native clang vector type (HIP's float4 is a struct and is rejected by the
// nontemporal builtins).
typedef float f4 __attribute__((ext_vector_type(4)));

#define COPY_THREADS 256
#define COPY_CHUNKS  8          // 8 x 16B per thread staged in LDS (32KB/block)
#define HEAD_DIM     128        // floats per row (from reference)
#define F4_PER_ROW   (HEAD_DIM / 4)

// ---------------------------------------------------------------------------
// Bulk copy: src[0..n4) f4 -> dst[0..n4), staged through LDS with the
// CDNA5 async-DMA path. Each block owns one contiguous tile of
// COPY_THREADS*COPY_CHUNKS f4 (grid-stride loop for safety).
// ---------------------------------------------------------------------------
__global__ __launch_bounds__(COPY_THREADS)
void async_copy_kernel(const f4* __restrict__ src,
                       f4* __restrict__ dst,
                       long n4) {
    __shared__ f4 stage[COPY_THREADS * COPY_CHUNKS];   // 32 KB of LDS

    const unsigned tid = threadIdx.x;
    // Byte address of this thread's chunk-0 LDS slot. Generic pointers to LDS
    // carry the LDS byte offset in their low 32 bits (ISA 10.2 aperture rules).
    const unsigned lds0 =
        (unsigned)(unsigned long long)(void*)&stage[tid];

    const long tile   = (long)COPY_THREADS * COPY_CHUNKS;
    const long stride = (long)gridDim.x * tile;

    for (long base = (long)blockIdx.x * tile + tid; base < n4; base += stride) {
        // Phase 1: issue up to COPY_CHUNKS async HBM->LDS b128 transfers.
#pragma unroll
        for (int c = 0; c < COPY_CHUNKS; ++c) {
            long idx = base + (long)c * COPY_THREADS;
            if (idx < n4) {
                const f4* g = src + idx;
                unsigned l = lds0 + (unsigned)(c * COPY_THREADS * sizeof(f4));
                asm volatile("global_load_async_to_lds_b128 %0, %1, off"
                             :: "v"(l), "v"(g) : "memory");
            }
        }
        asm volatile("s_wait_asynccnt 0" ::: "memory");

        // Phase 2: issue async LDS->HBM b128 transfers.
#pragma unroll
        for (int c = 0; c < COPY_CHUNKS; ++c) {
            long idx = base + (long)c * COPY_THREADS;
            if (idx < n4) {
                f4* g = dst + idx;
                unsigned l = lds0 + (unsigned)(c * COPY_THREADS * sizeof(f4));
                asm volatile("global_store_async_from_lds_b128 %0, %1, off"
                             :: "v"(g), "v"(l) : "memory");
            }
        }
        // Drain before LDS slots are reused (and before wave retires).
        asm volatile("s_wait_asynccnt 0" ::: "memory");
    }
}

// ---------------------------------------------------------------------------
// Scatter: one wave32 per (head, token) row; lane i moves f4 i of the
// 512B row. Page-table translation is computed per wave from device-resident
// seq_id / page_size / positions / page_table.
//   out_k[(h*n_slots + phys)*128 .. +128) = k[(h*T + t)*128 .. +128)
// ---------------------------------------------------------------------------
__global__ __launch_bounds__(256)
void scatter_kv(const f4* __restrict__ k,
                const f4* __restrict__ v,
                const int* __restrict__ page_table,
                const int* __restrict__ positions,
                const int* __restrict__ seq_id_p,
                const int* __restrict__ page_size_p,
                f4* __restrict__ out,
                long n4_half,      // f4 elements per cache half
                int rows,          // H*T
                int T,             // tokens
                int n_slots) {     // physical slots per head
    const int lane = threadIdx.x & 31;
    const int wave = threadIdx.x >> 5;
    const int row  = blockIdx.x * (blockDim.x >> 5) + wave;
    if (row >= rows) return;

    const int h = row / T;
    const int t = row - h * T;

    const int ps      = page_size_p[0];
    const int sid     = seq_id_p[0];
    const int row_len = n_slots / ps;         // page_table row length (N_PAGES)

    const int pos  = positions[t];
    const int lp   = pos / ps;                // logical page
    const int off  = pos - lp * ps;           // offset within page
    const int phys = page_table[(long)sid * row_len + lp] * ps + off;

    const long src_idx = (long)row * F4_PER_ROW + lane;
    const long dst_idx = ((long)h * n_slots + phys) * F4_PER_ROW + lane;

    const f4* srcp;
    f4*       dstp;
    if (blockIdx.y == 0) { srcp = k; dstp = out; }
    else                 { srcp = v; dstp = out + n4_half; }

    // Streamed once, never re-read by this dispatch -> non-temporal policy.
    f4 val = __builtin_nontemporal_load(&srcp[src_idx]);
    __builtin_nontemporal_store(val, &dstp[dst_idx]);
}

extern "C" void kernel_launch(void* const* d_in, const int* in_sizes, int n_in,
                              void* d_out, int out_size, void* d_ws, size_t ws_size,
                              hipStream_t stream) {
    const f4* k_cache = (const f4*)d_in[0];
    const f4* v_cache = (const f4*)d_in[1];
    const f4* k       = (const f4*)d_in[2];
    const f4* v       = (const f4*)d_in[3];
    const int* page_table = (const int*)d_in[4];
    const int* positions  = (const int*)d_in[5];
    const int* seq_id_p   = (const int*)d_in[6];
    const int* page_size_p= (const int*)d_in[7];
    f4* out = (f4*)d_out;

    const long cache_floats = (long)in_sizes[0];          // H * n_slots * 128
    const long n4_half      = cache_floats / 4;           // f4 per half
    const int  T            = in_sizes[5];                // tokens written
    const int  rows         = in_sizes[2] / HEAD_DIM;     // H * T
    const int  H            = rows / T;
    const int  n_slots      = (int)(cache_floats / ((long)H * HEAD_DIM));

    // 1) Bulk-copy both caches into the output via the async-DMA path.
    const long tile = (long)COPY_THREADS * COPY_CHUNKS;   // f4 per block
    const int  copy_blocks = (int)((n4_half + tile - 1) / tile);
    async_copy_kernel<<<copy_blocks, COPY_THREADS, 0, stream>>>(k_cache, out, n4_half);
    async_copy_kernel<<<copy_blocks, COPY_THREADS, 0, stream>>>(v_cache, out + n4_half, n4_half);

    // 2) Scatter the new K/V rows on top (stream order guarantees they win).
    dim3 grid((rows + 7) / 8, 2, 1);                      // 8 waves/block, y: k vs v
    scatter_kv<<<grid, 256, 0, stream>>>(k, v, page_table, positions,
                                         seq_id_p, page_size_p,
                                         out, n4_half, rows, T, n_slots);
}